// compute_A_9225589752483
// MI455X (gfx1250) — compile-verified
//
#include <hip/hip_runtime.h>
#include <cstddef>

// ---------------- problem constants ----------------
#define NIM   320
#define GRID  640
#define JW    6
#define KPTS  102400
#define NCOIL 8
#define ALPHA_KB 14.04f           // 2.34 * 6
#define ALPHA2_KB 197.1216f       // ALPHA^2
#define TRAJ_SCALE 101.85916357881302f   // G / (2*pi)
#define PI_F 3.14159265358979323846f
#define TWO_PI_F 6.28318530717958647692f

typedef float v2f __attribute__((ext_vector_type(2)));
typedef float v8f __attribute__((ext_vector_type(8)));

#if defined(__HIP_DEVICE_COMPILE__) && __has_builtin(__builtin_amdgcn_wmma_f32_16x16x4_f32)
#define HAVE_WMMA_F32X4 1
#else
#define HAVE_WMMA_F32X4 0
#endif

#if HAVE_WMMA_F32X4
static __device__ __forceinline__ v8f wmma4(v2f a, v2f b, v8f c) {
  // D(16x16) += A(16x4) * B(4x16), fp32, wave32
  return __builtin_amdgcn_wmma_f32_16x16x4_f32(false, a, false, b, (short)0, c,
                                               false, false);
}
#endif

// ---------------- kernel 1: coil mult + apodization + zero-pad ----------------
static __device__ __forceinline__ float apod_inv_1d(int i) {
  // 1/ax[i] = sinh(s)/s with s = sqrt(alpha^2 - (pi*J*x)^2), x=(i-N/2)/G
  float xv = ((float)i - 160.0f) * (1.0f / 640.0f);
  float q = PI_F * 6.0f * xv;
  float s = sqrtf(ALPHA2_KB - q * q);
  return sinhf(s) / s;
}

__global__ __launch_bounds__(256) void prep_kernel(
    const float* __restrict__ imr, const float* __restrict__ imi,
    const float* __restrict__ csr, const float* __restrict__ csi,
    float2* __restrict__ grid) {
  int p = blockIdx.x * 256 + threadIdx.x;          // 0 .. 640*640-1
  if (p >= GRID * GRID) return;
  int g0 = p / GRID, g1 = p % GRID;
  if (g0 < NIM && g1 < NIM) {
    // fold apodization and the fft2 "ortho" 1/G normalization here
    float fac = apod_inv_1d(g0) * apod_inv_1d(g1) * (1.0f / 640.0f);
    int ip = g0 * NIM + g1;
    float ar = imr[ip] * fac, ai = imi[ip] * fac;
#pragma unroll
    for (int c = 0; c < NCOIL; ++c) {
      float cr = csr[(size_t)c * (NIM * NIM) + ip];
      float ci = csi[(size_t)c * (NIM * NIM) + ip];
      float2 v;
      v.x = ar * cr - ai * ci;
      v.y = ar * ci + ai * cr;
      grid[(size_t)c * (GRID * GRID) + p] = v;
    }
  } else {
#pragma unroll
    for (int c = 0; c < NCOIL; ++c)
      grid[(size_t)c * (GRID * GRID) + p] = make_float2(0.f, 0.f);
  }
}

// ---------------- kernel 2: 640-pt row FFT (four-step 16x40), transposed write --------
// One wave32 per row. Output written transposed so a second identical pass
// completes the 2D FFT and restores orientation.
#define FFT_WAVES 8
#define LDS_FLOATS (1792 + FFT_WAVES * 2816)   // 24320 floats = 97280 B

__global__ __launch_bounds__(256) void fft640_kernel(
    const float2* __restrict__ src, float2* __restrict__ dst) {
  extern __shared__ float lds[];
  const int tid = threadIdx.x;
  const int lane = tid & 31;
  const int wv = tid >> 5;
  const int half = lane >> 4;     // 0/1: which 16-lane half of the wave
  const int l16 = lane & 15;

  float* F16R = lds;              // [16][16]
  float* F16I = lds + 256;
  float* TWR  = lds + 512;        // [16][40] twiddles exp(-2pi i k1 n2 / 640)
  float* TWI  = lds + 512 + 640;
  float* wb   = lds + 1792 + wv * 2816;
  float* XR = wb;                 // [16][40] row data, X[n1][n2] = x[40 n1 + n2]
  float* XI = wb + 640;
  float* YR = wb + 1280;          // [16][48] stage-1 output (padded cols)
  float* YI = wb + 1280 + 768;

  // --- build constant tables (exact integer-mod angles for precision) ---
  for (int idx = tid; idx < 256; idx += 256) {
    int k1 = idx >> 4, n1 = idx & 15;
    int m = (k1 * n1) & 15;
    float s, c;
    __sincosf(-(float)m * (TWO_PI_F / 16.0f), &s, &c);
    F16R[idx] = c; F16I[idx] = s;
  }
  for (int idx = tid; idx < 640; idx += 256) {
    int k1 = idx / 40, n2 = idx % 40;
    int m = (k1 * n2) % 640;
    float s, c;
    __sincosf(-(float)m * (TWO_PI_F / 640.0f), &s, &c);
    TWR[idx] = c; TWI[idx] = s;
  }
  __syncthreads();

  const int row  = blockIdx.x * FFT_WAVES + wv;   // grid = 640 blocks -> 5120 rows
  const int coil = row / GRID;
  const int r    = row % GRID;

  // stage row into LDS (coalesced)
  const float2* srow = src + (size_t)coil * (GRID * GRID) + (size_t)r * GRID;
  for (int e = lane; e < GRID; e += 32) {
    float2 v = srow[e];
    XR[e] = v.x; XI[e] = v.y;
  }
  __syncthreads();

  // ---------------- stage 1: Y(16x40) = F16(16x16) * X(16x40), then twiddle ---------
  for (int nt = 0; nt < 3; ++nt) {
    const int n2 = nt * 16 + l16;
    const bool inr = (n2 < 40);
    float accR[8], accI[8];
#if HAVE_WMMA_F32X4
    v8f yrv = {}; v8f yiv = {};
    for (int kk = 0; kk < 4; ++kk) {
      const int K0 = kk * 4 + 2 * half, K1 = K0 + 1;
      v2f afr, afi, anfi, bxr, bxi;
      afr.x = F16R[l16 * 16 + K0]; afr.y = F16R[l16 * 16 + K1];
      afi.x = F16I[l16 * 16 + K0]; afi.y = F16I[l16 * 16 + K1];
      anfi.x = -afi.x; anfi.y = -afi.y;
      bxr.x = inr ? XR[K0 * 40 + n2] : 0.f;  bxr.y = inr ? XR[K1 * 40 + n2] : 0.f;
      bxi.x = inr ? XI[K0 * 40 + n2] : 0.f;  bxi.y = inr ? XI[K1 * 40 + n2] : 0.f;
      yrv = wmma4(afr,  bxr, yrv);           // + Fr*Xr
      yrv = wmma4(anfi, bxi, yrv);           // - Fi*Xi
      yiv = wmma4(afr,  bxi, yiv);           // + Fr*Xi
      yiv = wmma4(afi,  bxr, yiv);           // + Fi*Xr
    }
#pragma unroll
    for (int rr = 0; rr < 8; ++rr) { accR[rr] = yrv[rr]; accI[rr] = yiv[rr]; }
#else
#pragma unroll
    for (int rr = 0; rr < 8; ++rr) { accR[rr] = 0.f; accI[rr] = 0.f; }
    for (int rr = 0; rr < 8; ++rr) {
      const int M = rr + 8 * half;
      for (int K = 0; K < 16; ++K) {
        float fr = F16R[M * 16 + K], fi = F16I[M * 16 + K];
        float xr = inr ? XR[K * 40 + n2] : 0.f;
        float xi = inr ? XI[K * 40 + n2] : 0.f;
        accR[rr] += fr * xr - fi * xi;
        accI[rr] += fr * xi + fi * xr;
      }
    }
#endif
#pragma unroll
    for (int rr = 0; rr < 8; ++rr) {
      const int M = rr + 8 * half;
      const int N = nt * 16 + l16;
      if (N < 40) {
        float tr = TWR[M * 40 + N], ti = TWI[M * 40 + N];
        YR[M * 48 + N] = accR[rr] * tr - accI[rr] * ti;
        YI[M * 48 + N] = accR[rr] * ti + accI[rr] * tr;
      } else {
        YR[M * 48 + N] = 0.f; YI[M * 48 + N] = 0.f;
      }
    }
  }
  __syncthreads();

  // ---------------- stage 2: Z(16x40) = Y(16x40) * F40(40x40); write transposed -----
  float2* drow = dst + (size_t)coil * (GRID * GRID);
  for (int nt = 0; nt < 3; ++nt) {
    const int k2 = nt * 16 + l16;
    float accR[8], accI[8];
#if HAVE_WMMA_F32X4
    v8f zrv = {}; v8f ziv = {};
    for (int kk = 0; kk < 10; ++kk) {
      const int K0 = kk * 4 + 2 * half, K1 = K0 + 1;
      v2f ayr, ayi, anyi, bfr, bfi;
      ayr.x = YR[l16 * 48 + K0]; ayr.y = YR[l16 * 48 + K1];
      ayi.x = YI[l16 * 48 + K0]; ayi.y = YI[l16 * 48 + K1];
      anyi.x = -ayi.x; anyi.y = -ayi.y;
      int m0 = (K0 * k2) % 40, m1 = (K1 * k2) % 40;
      float s0, c0, s1, c1;
      __sincosf(-(float)m0 * (TWO_PI_F / 40.0f), &s0, &c0);
      __sincosf(-(float)m1 * (TWO_PI_F / 40.0f), &s1, &c1);
      bfr.x = c0; bfr.y = c1;                // Re F40
      bfi.x = s0; bfi.y = s1;                // Im F40
      zrv = wmma4(ayr,  bfr, zrv);
      zrv = wmma4(anyi, bfi, zrv);
      ziv = wmma4(ayr,  bfi, ziv);
      ziv = wmma4(ayi,  bfr, ziv);
    }
#pragma unroll
    for (int rr = 0; rr < 8; ++rr) { accR[rr] = zrv[rr]; accI[rr] = ziv[rr]; }
#else
#pragma unroll
    for (int rr = 0; rr < 8; ++rr) { accR[rr] = 0.f; accI[rr] = 0.f; }
    for (int rr = 0; rr < 8; ++rr) {
      const int M = rr + 8 * half;
      for (int K = 0; K < 40; ++K) {
        int m = (K * k2) % 40;
        float s, c;
        __sincosf(-(float)m * (TWO_PI_F / 40.0f), &s, &c);
        float yr = YR[M * 48 + K], yi = YI[M * 48 + K];
        accR[rr] += yr * c - yi * s;
        accI[rr] += yr * s + yi * c;
      }
    }
#endif
    if (k2 < 40) {
#pragma unroll
      for (int rr = 0; rr < 8; ++rr) {
        const int k1 = rr + 8 * half;
        const int e = k1 + 16 * k2;                 // output frequency index
        drow[(size_t)e * GRID + r] = make_float2(accR[rr], accI[rr]);  // transposed
      }
    }
  }
}

// ---------------- kernel 3: KB interpolation + mask ----------------
static __device__ __forceinline__ float i0f_dev(float x) {
  if (x < 3.75f) {
    float t = x * x * (1.0f / 14.0625f);
    return 1.0f + t * (3.5156229f + t * (3.0899424f + t * (1.2067492f +
               t * (0.2659732f + t * (0.0360768f + t * 0.0045813f)))));
  }
  float ti = 3.75f / x;
  float p = 0.39894228f + ti * (0.01328592f + ti * (0.00225319f + ti * (-0.00157565f +
            ti * (0.00916281f + ti * (-0.02057706f + ti * (0.02635537f +
            ti * (-0.01647633f + ti * 0.00392377f)))))));
  return expf(x) * p / sqrtf(x);
}

static __device__ __forceinline__ float kb_weight(float u, float inv_i0a) {
  if (fabsf(u) >= 3.0f) return 0.0f;
  float q = u * (1.0f / 3.0f);
  float t = fmaxf(1.0f - q * q, 0.0f);
  return i0f_dev(ALPHA_KB * sqrtf(t)) * inv_i0a;
}

__global__ __launch_bounds__(256) void interp_kernel(
    const float2* __restrict__ grid, const float* __restrict__ ktraj,
    const float* __restrict__ kmask, float* __restrict__ out, float inv_i0a) {
  int k = blockIdx.x * 256 + threadIdx.x;
  if (k >= KPTS) return;
  float tx = ktraj[k] * TRAJ_SCALE;           // axis 0 (rows of grid)
  float ty = ktraj[KPTS + k] * TRAJ_SCALE;    // axis 1 (cols of grid)
  int bx = (int)floorf(tx) - 2;
  int by = (int)floorf(ty) - 2;

  float wx[6], wy[6];
  int ix[6], iy[6];
#pragma unroll
  for (int j = 0; j < 6; ++j) {
    wx[j] = kb_weight(tx - (float)(bx + j), inv_i0a);
    wy[j] = kb_weight(ty - (float)(by + j), inv_i0a);
    int vx = bx + j; vx += (vx < 0) ? GRID : 0; vx -= (vx >= GRID) ? GRID : 0;
    int vy = by + j; vy += (vy < 0) ? GRID : 0; vy -= (vy >= GRID) ? GRID : 0;
    ix[j] = vx; iy[j] = vy;
  }

  float accr[NCOIL], acci[NCOIL];
#pragma unroll
  for (int c = 0; c < NCOIL; ++c) { accr[c] = 0.f; acci[c] = 0.f; }

#pragma unroll
  for (int i = 0; i < 6; ++i) {
    const int rb = ix[i] * GRID;
    const float wxi = wx[i];
#pragma unroll
    for (int j = 0; j < 6; ++j) {
      const float w = wxi * wy[j];
      const float2* p = grid + rb + iy[j];
#pragma unroll
      for (int c = 0; c < NCOIL; ++c) {
        float2 v = p[(size_t)c * (GRID * GRID)];
        accr[c] += w * v.x;
        acci[c] += w * v.y;
      }
    }
  }
  const float m = kmask[k];
#pragma unroll
  for (int c = 0; c < NCOIL; ++c) {
    size_t o = ((size_t)c * KPTS + (size_t)k) * 2;
    out[o]     = accr[c] * m;
    out[o + 1] = acci[c] * m;
  }
}

// ---------------- host entry ----------------
extern "C" void kernel_launch(void* const* d_in, const int* in_sizes, int n_in,
                              void* d_out, int out_size, void* d_ws, size_t ws_size,
                              hipStream_t stream) {
  (void)in_sizes; (void)n_in; (void)out_size; (void)ws_size;
  const float* imr = (const float*)d_in[0];
  const float* imi = (const float*)d_in[1];
  const float* csr = (const float*)d_in[2];
  const float* csi = (const float*)d_in[3];
  const float* ktr = (const float*)d_in[4];
  const float* kms = (const float*)d_in[5];

  float2* bufA = (float2*)d_ws;                         // 8*640*640 float2 = 26.2 MB
  float2* bufB = bufA + (size_t)NCOIL * GRID * GRID;    // second 26.2 MB

  // host-side I0(alpha) (deterministic series)
  double hh = (14.04 * 0.5) * (14.04 * 0.5);
  double term = 1.0, sum = 1.0;
  for (int m = 1; m < 128; ++m) {
    term *= hh / ((double)m * (double)m);
    sum += term;
    if (term < sum * 1e-18) break;
  }
  float inv_i0a = (float)(1.0 / sum);

  prep_kernel<<<(GRID * GRID + 255) / 256, 256, 0, stream>>>(imr, imi, csr, csi, bufA);
  fft640_kernel<<<GRID, 256, LDS_FLOATS * sizeof(float), stream>>>(bufA, bufB); // axis1 FFT, write ^T
  fft640_kernel<<<GRID, 256, LDS_FLOATS * sizeof(float), stream>>>(bufB, bufA); // axis0 FFT, write ^T
  interp_kernel<<<(KPTS + 255) / 256, 256, 0, stream>>>(bufA, ktr, kms, (float*)d_out, inv_i0a);
}